// EFGPND_27358941676004
// MI455X (gfx1250) — compile-verified
//
#include <hip/hip_runtime.h>
#include <hip/hip_bf16.h>
#include <math.h>

// ---------------------------------------------------------------------------
// EFGP 2D: equispaced-Fourier GP regression.
//   MTOT=65 modes/dim (M=4225), lengthscale 0.1, outputscale 1, sigma^2 0.05,
//   30 CG iterations, N_train=100000, N_test=50000.
// Pipeline: minmax -> (h, ws grid) -> v (129x129, WMMA f32 GEMM over N)
//        -> rhs (65x65, WMMA f32 GEMM over N, weighted by y)
//        -> single-block CG with direct Toeplitz convolution
//        -> type-2 NUDFT at test points via phase recurrences.
// ---------------------------------------------------------------------------

typedef __attribute__((ext_vector_type(2))) float v2f;
typedef __attribute__((ext_vector_type(8))) float v8f;

#define MG   65
#define MM   4225          // 65*65
#define K2   129           // 2*65-1
#define TWOPI_F 6.28318530717958647692f
#define PI_F    3.14159265358979323846f
#define SIGMASQ_F 0.05f
#define LSQ_F (0.1f * 0.1f)
#define CG_ITERS 30

// ---- ordered-uint encoding of float for deterministic atomic min/max ------
__device__ __forceinline__ unsigned efgp_f2o(float f) {
    unsigned u = __float_as_uint(f);
    return (u & 0x80000000u) ? ~u : (u | 0x80000000u);
}
__device__ __forceinline__ float efgp_o2f(unsigned o) {
    unsigned u = (o & 0x80000000u) ? (o & 0x7FFFFFFFu) : ~o;
    return __uint_as_float(u);
}

// ---- init the 4 minmax slots ----------------------------------------------
__global__ void efgp_init_mm(unsigned* mm) {
    int t = threadIdx.x;
    if (t < 4) mm[t] = (t & 1) ? 0u : 0xFFFFFFFFu;   // even: min slot, odd: max slot
}

// ---- deterministic min/max over x and x_new (atomics on ordered uints) ----
__global__ void efgp_minmax(const float* __restrict__ x, int ntr,
                            const float* __restrict__ xn, int nte,
                            unsigned* mm) {
    int total = ntr + nte;
    for (int i = blockIdx.x * blockDim.x + threadIdx.x; i < total;
         i += gridDim.x * blockDim.x) {
        float a, b;
        if (i < ntr) { a = x[2 * i]; b = x[2 * i + 1]; }
        else         { a = xn[2 * (i - ntr)]; b = xn[2 * (i - ntr) + 1]; }
        atomicMin(&mm[0], efgp_f2o(a));
        atomicMax(&mm[1], efgp_f2o(a));
        atomicMin(&mm[2], efgp_f2o(b));
        atomicMax(&mm[3], efgp_f2o(b));
    }
}

// ---- compute h and the spectral weight grid ws (65x65, real) --------------
__global__ void efgp_prep(const unsigned* __restrict__ mm,
                          float* __restrict__ hslot, float* __restrict__ wsg) {
    float L = fmaxf(efgp_o2f(mm[1]) - efgp_o2f(mm[0]),
                    efgp_o2f(mm[3]) - efgp_o2f(mm[2]));
    if (L <= 1e-9f) L = 1.0f;
    float h = 1.0f / L;
    int i = blockIdx.x * blockDim.x + threadIdx.x;
    if (i == 0) hslot[0] = h;
    if (i < MM) {
        int a = i / MG, b = i - a * MG;
        float xi0 = h * (float)(a - 32);
        float xi1 = h * (float)(b - 32);
        // S(s) = sigma_f^2 (2 pi l^2)^(D/2) exp(-2 pi^2 l^2 |s|^2), D=2
        float spec = (TWOPI_F * LSQ_F) *
                     __expf(-2.0f * PI_F * PI_F * LSQ_F * (xi0 * xi0 + xi1 * xi1));
        wsg[i] = sqrtf(spec * h * h);
    }
}

// ---- one K=4 chunk: 4 sincos + 4 f32 WMMAs --------------------------------
// phA = 2*pi*h*offA (row side, dim 0), phB = 2*pi*h*offB (col side, dim 1).
// Complex product exp(-i thA) * exp(-i thB) accumulated as:
//   Re += cosA*cosB + (+sinA)*(-sinB)    Im += cosA*(-sinB) + (-sinA)*cosB
__device__ __forceinline__ void efgp_chunk(float phA, float phB,
                                           float x0a, float x1a, float w0,
                                           float x0b, float x1b, float w1,
                                           v8f& accRe, v8f& accIm) {
    float cA0, sA0, cB0, sB0, cA1, sA1, cB1, sB1;
    __sincosf(phA * x0a, &sA0, &cA0);
    __sincosf(phB * x1a, &sB0, &cB0);
    __sincosf(phA * x0b, &sA1, &cA1);
    __sincosf(phB * x1b, &sB1, &cB1);
    v2f Ac  = {  w0 * cA0,  w1 * cA1 };   // weighted cos(thetaA)
    v2f Ams = { -w0 * sA0, -w1 * sA1 };   // weighted -sin(thetaA)
    v2f Aps = {  w0 * sA0,  w1 * sA1 };   // weighted +sin(thetaA)
    v2f Bc  = {  cB0,  cB1 };             // cos(thetaB)
    v2f Bms = { -sB0, -sB1 };             // -sin(thetaB)
    accRe = __builtin_amdgcn_wmma_f32_16x16x4_f32(false, Ac,  false, Bc,
                                                  (short)0, accRe, false, false);
    accRe = __builtin_amdgcn_wmma_f32_16x16x4_f32(false, Aps, false, Bms,
                                                  (short)0, accRe, false, false);
    accIm = __builtin_amdgcn_wmma_f32_16x16x4_f32(false, Ac,  false, Bms,
                                                  (short)0, accIm, false, false);
    accIm = __builtin_amdgcn_wmma_f32_16x16x4_f32(false, Ams, false, Bc,
                                                  (short)0, accIm, false, false);
}

// ---- WMMA type-1 NUDFT tile kernel ----------------------------------------
// out[a,b] = sum_n w_n exp(-i*2pi*h*off_a*x_n0) exp(-i*2pi*h*off_b*x_n1)
// One wave per 16x16 output tile; loops over all N in guard-free K=4 chunks
// (single guarded tail chunk), so no cross-block accumulation -> deterministic.
template <bool HASW>
__global__ __launch_bounds__(32)
void efgp_nudft1_tile(const float* __restrict__ x, const float* __restrict__ wgt,
                      int n, const float* __restrict__ hslot,
                      int nOff, int offStart, int T,
                      float* __restrict__ outRe, float* __restrict__ outIm) {
    const float h = hslot[0];
    const int t0 = blockIdx.x / T;
    const int t1 = blockIdx.x % T;
    const int lane = threadIdx.x;
    const int m = lane & 15;               // A row / B col within tile
    const int kbase = (lane >> 4) << 1;    // 0 or 2
    const float phA = TWOPI_F * h * (float)(offStart + t0 * 16 + m);
    const float phB = TWOPI_F * h * (float)(offStart + t1 * 16 + m);

    v8f accRe = {};
    v8f accIm = {};

    const float2* __restrict__ xv = (const float2*)x;
    const int nAligned = n & ~3;

    // guard-free main loop
    for (int nn = 0; nn < nAligned; nn += 4) {
        int p0 = nn + kbase;
        float2 xy0 = xv[p0];
        float2 xy1 = xv[p0 + 1];
        float w0 = HASW ? wgt[p0] : 1.0f;
        float w1 = HASW ? wgt[p0 + 1] : 1.0f;
        efgp_chunk(phA, phB, xy0.x, xy0.y, w0, xy1.x, xy1.y, w1, accRe, accIm);
    }
    // guarded tail chunk (at most one)
    if (nAligned < n) {
        int p0 = nAligned + kbase;
        int p1 = p0 + 1;
        int c0 = (p0 < n) ? p0 : 0;
        int c1 = (p1 < n) ? p1 : 0;
        float2 xy0 = xv[c0];
        float2 xy1 = xv[c1];
        float w0 = (p0 < n) ? (HASW ? wgt[c0] : 1.0f) : 0.0f;
        float w1 = (p1 < n) ? (HASW ? wgt[c1] : 1.0f) : 0.0f;
        efgp_chunk(phA, phB, xy0.x, xy0.y, w0, xy1.x, xy1.y, w1, accRe, accIm);
    }

    // D layout: VGPR i -> row i (lanes 0-15) or row i+8 (lanes 16-31), col=lane&15
    const int col = lane & 15;
    const int b_off = t1 * 16 + col;
    for (int i2 = 0; i2 < 8; ++i2) {
        int rowM = (lane < 16) ? i2 : (8 + i2);
        int a_off = t0 * 16 + rowM;
        if (a_off < nOff && b_off < nOff) {
            outRe[a_off * nOff + b_off] = accRe[i2];
            outIm[a_off * nOff + b_off] = accIm[i2];
        }
    }
}

// ---- single-block CG (deterministic LDS tree reductions) ------------------
// Solves (W F^H F W + sigma^2 I) beta = W F^H y.  Toeplitz apply is the direct
// 2D linear convolution with v (exactly equal to the reference's padded FFT).
__global__ __launch_bounds__(1024)
void efgp_cg(const float* __restrict__ vRe, const float* __restrict__ vIm,
             const float* __restrict__ rhsRe, const float* __restrict__ rhsIm,
             const float* __restrict__ wsg,
             float* __restrict__ xkRe, float* __restrict__ xkIm,
             float* __restrict__ rRe,  float* __restrict__ rIm,
             float* __restrict__ pRe,  float* __restrict__ pIm,
             float* __restrict__ wbRe, float* __restrict__ wbIm,
             float* __restrict__ ApRe, float* __restrict__ ApIm,
             float* __restrict__ fkRe, float* __restrict__ fkIm) {
    __shared__ float sred[1024];
    __shared__ float sc[3];   // 0: rs, 1: alpha, 2: beta
    const int tid = threadIdx.x;

    // init: b = ws .* rhs_raw ; x=0 ; r=p=b ; rs=||b||^2
    float loc = 0.0f;
    for (int i = tid; i < MM; i += 1024) {
        float br = wsg[i] * rhsRe[i], bi = wsg[i] * rhsIm[i];
        xkRe[i] = 0.0f; xkIm[i] = 0.0f;
        rRe[i] = br; rIm[i] = bi;
        pRe[i] = br; pIm[i] = bi;
        loc += br * br + bi * bi;
    }
    sred[tid] = loc; __syncthreads();
    for (int s = 512; s > 0; s >>= 1) {
        if (tid < s) sred[tid] += sred[tid + s];
        __syncthreads();
    }
    if (tid == 0) sc[0] = sred[0];
    __syncthreads();

    for (int it = 0; it < CG_ITERS; ++it) {
        // wb = ws .* p
        for (int i = tid; i < MM; i += 1024) {
            wbRe[i] = wsg[i] * pRe[i];
            wbIm[i] = wsg[i] * pIm[i];
        }
        __syncthreads();

        // Ap = ws .* conv(v, wb) + sigma^2 * p ; pAp = Re<p,Ap>
        float lpap = 0.0f;
        for (int i = tid; i < MM; i += 1024) {
            int a = i / MG, b = i - a * MG;
            float tr = 0.0f, ti = 0.0f;
            for (int ap = 0; ap < MG; ++ap) {
                const float* vr_row = vRe + (a - ap + 64) * K2 + (b + 64);
                const float* vi_row = vIm + (a - ap + 64) * K2 + (b + 64);
                const float* ur_row = wbRe + ap * MG;
                const float* ui_row = wbIm + ap * MG;
                for (int bp = 0; bp < MG; ++bp) {
                    float vr = vr_row[-bp], vi = vi_row[-bp];
                    float ur = ur_row[bp],  ui = ui_row[bp];
                    tr += vr * ur - vi * ui;
                    ti += vr * ui + vi * ur;
                }
            }
            float ar = wsg[i] * tr + SIGMASQ_F * pRe[i];
            float ai = wsg[i] * ti + SIGMASQ_F * pIm[i];
            ApRe[i] = ar; ApIm[i] = ai;
            lpap += pRe[i] * ar + pIm[i] * ai;
        }
        sred[tid] = lpap; __syncthreads();
        for (int s = 512; s > 0; s >>= 1) {
            if (tid < s) sred[tid] += sred[tid + s];
            __syncthreads();
        }
        if (tid == 0) sc[1] = sc[0] / sred[0];   // alpha = rs / pAp
        __syncthreads();
        float alpha = sc[1];

        // x += alpha p ; r -= alpha Ap ; rs_new = ||r||^2
        float lrs = 0.0f;
        for (int i = tid; i < MM; i += 1024) {
            xkRe[i] += alpha * pRe[i];
            xkIm[i] += alpha * pIm[i];
            float nr = rRe[i] - alpha * ApRe[i];
            float ni = rIm[i] - alpha * ApIm[i];
            rRe[i] = nr; rIm[i] = ni;
            lrs += nr * nr + ni * ni;
        }
        sred[tid] = lrs; __syncthreads();
        for (int s = 512; s > 0; s >>= 1) {
            if (tid < s) sred[tid] += sred[tid + s];
            __syncthreads();
        }
        if (tid == 0) { sc[2] = sred[0] / sc[0]; sc[0] = sred[0]; }
        __syncthreads();
        float bet = sc[2];

        // p = r + beta p
        for (int i = tid; i < MM; i += 1024) {
            pRe[i] = rRe[i] + bet * pRe[i];
            pIm[i] = rIm[i] + bet * pIm[i];
        }
        __syncthreads();
    }

    // fk = ws .* beta
    for (int i = tid; i < MM; i += 1024) {
        fkRe[i] = wsg[i] * xkRe[i];
        fkIm[i] = wsg[i] * xkIm[i];
    }
}

// ---- type-2 NUDFT at test points via phase recurrences --------------------
// yhat_n = Re( sum_{a,b} fk[a,b] e^{+i th_a(x0)} e^{+i th_b(x1)} ), k=-32..32
__global__ void efgp_type2(const float* __restrict__ xn, int n,
                           const float* __restrict__ hslot,
                           const float* __restrict__ fkRe,
                           const float* __restrict__ fkIm,
                           float* __restrict__ out) {
    int i = blockIdx.x * blockDim.x + threadIdx.x;
    if (i >= n) return;
    float h = hslot[0];
    float p0 = TWOPI_F * h * xn[2 * i];
    float p1 = TWOPI_F * h * xn[2 * i + 1];
    float s, c;
    __sincosf(-32.0f * p0, &s, &c); float e0r = c, e0i = s;
    __sincosf(p0, &s, &c);          float w0r = c, w0i = s;
    __sincosf(-32.0f * p1, &s, &c); float e1sr = c, e1si = s;
    __sincosf(p1, &s, &c);          float w1r = c, w1i = s;

    float acc = 0.0f;
    for (int a = 0; a < MG; ++a) {
        float e1r = e1sr, e1i = e1si;
        float gr = 0.0f, gi = 0.0f;
        const float* fr = fkRe + a * MG;
        const float* fi = fkIm + a * MG;
        for (int b = 0; b < MG; ++b) {
            float kr = fr[b], ki = fi[b];
            gr += kr * e1r - ki * e1i;
            gi += kr * e1i + ki * e1r;
            float nr = e1r * w1r - e1i * w1i;
            e1i = e1r * w1i + e1i * w1r;
            e1r = nr;
        }
        acc += gr * e0r - gi * e0i;
        float nr = e0r * w0r - e0i * w0i;
        e0i = e0r * w0i + e0i * w0r;
        e0r = nr;
    }
    out[i] = acc;
}

// ---------------------------------------------------------------------------
extern "C" void kernel_launch(void* const* d_in, const int* in_sizes, int n_in,
                              void* d_out, int out_size, void* d_ws, size_t ws_size,
                              hipStream_t stream) {
    const float* x    = (const float*)d_in[0];   // [N_train, 2]
    const float* y    = (const float*)d_in[1];   // [N_train]
    const float* xnew = (const float*)d_in[2];   // [N_test, 2]
    const int ntr = in_sizes[0] / 2;
    const int nte = in_sizes[2] / 2;

    // workspace layout (floats)
    float* W = (float*)d_ws;
    unsigned* mm = (unsigned*)d_ws;     // [0..3] ordered-uint min/max slots
    float* hslot = W + 4;               // [1]
    float* wsg   = W + 8;               // 4225
    float* vRe   = wsg + MM;            // 129*129
    float* vIm   = vRe + K2 * K2;
    float* rhsRe = vIm + K2 * K2;       // 4225 (raw type-1, pre ws-scale)
    float* rhsIm = rhsRe + MM;
    float* q     = rhsIm + MM;
    float* xkRe = q;  q += MM;  float* xkIm = q;  q += MM;
    float* rRe  = q;  q += MM;  float* rIm  = q;  q += MM;
    float* pRe  = q;  q += MM;  float* pIm  = q;  q += MM;
    float* wbRe = q;  q += MM;  float* wbIm = q;  q += MM;
    float* ApRe = q;  q += MM;  float* ApIm = q;  q += MM;
    float* fkRe = q;  q += MM;  float* fkIm = q;  q += MM;

    // 1) min/max reduction (deterministic integer atomics)
    efgp_init_mm<<<1, 32, 0, stream>>>(mm);
    {
        int total = ntr + nte;
        int blocks = (total + 255) / 256;
        efgp_minmax<<<blocks, 256, 0, stream>>>(x, ntr, xnew, nte, mm);
    }

    // 2) h and spectral weights
    efgp_prep<<<(MM + 255) / 256, 256, 0, stream>>>(mm, hslot, wsg);

    // 3) Toeplitz vector v: 129x129, offsets -64..64  (T = ceil(129/16) = 9)
    efgp_nudft1_tile<false><<<9 * 9, 32, 0, stream>>>(x, nullptr, ntr, hslot,
                                                      K2, -64, 9, vRe, vIm);

    // 4) raw rhs: 65x65, offsets -32..32, weighted by y (T = ceil(65/16) = 5)
    efgp_nudft1_tile<true><<<5 * 5, 32, 0, stream>>>(x, y, ntr, hslot,
                                                     MG, -32, 5, rhsRe, rhsIm);

    // 5) CG solve (single block, internal iteration, writes fk = ws.*beta)
    efgp_cg<<<1, 1024, 0, stream>>>(vRe, vIm, rhsRe, rhsIm, wsg,
                                    xkRe, xkIm, rRe, rIm, pRe, pIm,
                                    wbRe, wbIm, ApRe, ApIm, fkRe, fkIm);

    // 6) posterior mean at test points
    efgp_type2<<<(nte + 255) / 256, 256, 0, stream>>>(xnew, nte, hslot,
                                                      fkRe, fkIm, (float*)d_out);
}